// LAB_42906723287350
// MI455X (gfx1250) — compile-verified
//
#include <hip/hip_runtime.h>
#include <math.h>

typedef __attribute__((ext_vector_type(2))) float v2f;
typedef __attribute__((ext_vector_type(4))) float v4f;
typedef __attribute__((ext_vector_type(8))) float v8f;

#define C_IN 64
#define K_IN 4
// ws layout: [0..255] probs (C x K row-major), [256..271] sigmoid(lut), [272..274] mix softmax

// ---------------- prologue: parameter preprocessing (1 block) ----------------
__global__ void LAB_42906723287350_prep(const float* __restrict__ cp,
                                        const float* __restrict__ lut,
                                        const float* __restrict__ mixp,
                                        float* __restrict__ ws) {
    __shared__ float scp[C_IN * K_IN];
    __shared__ float smax[K_IN];
    __shared__ float ssum[K_IN];
    const int t = threadIdx.x;  // 256 threads
    scp[t] = cp[t];
    __syncthreads();
    if (t < K_IN) {
        float m = -3.0e38f;
        for (int c = 0; c < C_IN; ++c) m = fmaxf(m, scp[c * K_IN + t]);
        float s = 0.0f;
        for (int c = 0; c < C_IN; ++c) s += __expf(scp[c * K_IN + t] - m);
        smax[t] = m;
        ssum[t] = s;
    }
    __syncthreads();
    {
        const int k = t & (K_IN - 1);
        ws[t] = __expf(scp[t] - smax[k]) / ssum[k];
    }
    if (t < 16) ws[256 + t] = 1.0f / (1.0f + __expf(-lut[t]));
    if (t == 0) {
        float m0 = fmaxf(mixp[0], fmaxf(mixp[1], mixp[2]));
        float e0 = __expf(mixp[0] - m0);
        float e1 = __expf(mixp[1] - m0);
        float e2 = __expf(mixp[2] - m0);
        float inv = 1.0f / (e0 + e1 + e2);
        ws[272] = e0 * inv;
        ws[273] = e1 * inv;
        ws[274] = e2 * inv;
    }
}

// ---------------- main streaming kernel: 256 threads = 8 waves, 256 rows/block ----------------
__global__ void LAB_42906723287350_main(const float* __restrict__ x,
                                        const float* __restrict__ ws,
                                        float* __restrict__ out) {
    __shared__ float s_probs[C_IN * K_IN];  // 256 floats
    __shared__ float s_lut[16];
    __shared__ float s_mix[3];
    __shared__ float s_sig[256 * 4];        // per-row s0..s3 for epilogue

    const int t = threadIdx.x;
    s_probs[t] = ws[t];
    if (t < 16) s_lut[t] = ws[256 + t];
    if (t < 3)  s_mix[t] = ws[272 + t];
    __syncthreads();

    const int wave = t >> 5;       // 0..7
    const int lane = t & 31;
    const int mrow = lane & 15;    // M index for A, N index for B/D
    const int half = lane >> 4;    // 0: K layout {0,1}, 1: K layout {2,3}

    const size_t blockRow0 = (size_t)blockIdx.x * 256;
    const size_t tileRow0  = blockRow0 + (size_t)wave * 32;

    // B fragments (probs padded to N=16 with zeros), hoisted to registers.
    // iter 2t : layout-K rows = actual cols {8t+4h, 8t+4h+1}   (h = half)
    // iter 2t+1: layout-K rows = actual cols {8t+4h+2, 8t+4h+3}
    float bx[16], by[16];
    const bool bn = (mrow < K_IN);
#pragma unroll
    for (int tt = 0; tt < 8; ++tt) {
        const int ce = 8 * tt + half * 4;
        bx[2 * tt]     = bn ? s_probs[(ce    ) * K_IN + mrow] : 0.0f;
        by[2 * tt]     = bn ? s_probs[(ce + 1) * K_IN + mrow] : 0.0f;
        bx[2 * tt + 1] = bn ? s_probs[(ce + 2) * K_IN + mrow] : 0.0f;
        by[2 * tt + 1] = bn ? s_probs[(ce + 3) * K_IN + mrow] : 0.0f;
    }

    v8f acc0 = {};
    v8f acc1 = {};

    const float* rowp0 = x + (tileRow0 + (size_t)mrow) * C_IN + half * 4;  // 16B aligned
    const float* rowp1 = rowp0 + 16 * C_IN;

#pragma unroll
    for (int tt = 0; tt < 8; ++tt) {
        v4f a0 = *(const v4f*)(rowp0 + 8 * tt);   // global_load_b128, imm offset
        v4f a1 = *(const v4f*)(rowp1 + 8 * tt);
        v2f ae0 = {a0.x, a0.y}, ao0 = {a0.z, a0.w};
        v2f ae1 = {a1.x, a1.y}, ao1 = {a1.z, a1.w};
        v2f be  = {bx[2 * tt],     by[2 * tt]};
        v2f bo  = {bx[2 * tt + 1], by[2 * tt + 1]};
        acc0 = __builtin_amdgcn_wmma_f32_16x16x4_f32(false, ae0, false, be,
                                                     (short)0, acc0, false, false);
        acc1 = __builtin_amdgcn_wmma_f32_16x16x4_f32(false, ae1, false, be,
                                                     (short)0, acc1, false, false);
        acc0 = __builtin_amdgcn_wmma_f32_16x16x4_f32(false, ao0, false, bo,
                                                     (short)0, acc0, false, false);
        acc1 = __builtin_amdgcn_wmma_f32_16x16x4_f32(false, ao1, false, bo,
                                                     (short)0, acc1, false, false);
    }

    // Scatter the 4 valid D columns to LDS: D layout VGPR v -> (M = v + 8*half, N = lane&15)
    float* sbase = s_sig + wave * 32 * 4;
    if (mrow < 4) {
#pragma unroll
        for (int v = 0; v < 8; ++v) {
            const int r = v + half * 8;
            sbase[r * 4 + mrow]        = acc0[v];
            sbase[(r + 16) * 4 + mrow] = acc1[v];
        }
    }
    __syncthreads();

    // Epilogue: one row per thread
    v4f sv = *(const v4f*)(s_sig + t * 4);
    const float s0 = sv.x, s1 = sv.y, s2 = sv.z, s3 = sv.w;

    const float add = s0 + s1 + s2;
    const float a1  = fminf(fmaxf(add - 2.0f, 0.0f), 1.0f);
    const float a2  = (add >= 2.0f) ? 1.0f : 0.0f;

    // Multilinear contraction over sigmoid(lut): index value 0 -> weight s, 1 -> (1-s)
    float l3[8];
#pragma unroll
    for (int q = 0; q < 8; ++q)
        l3[q] = s3 * s_lut[2 * q] + (1.0f - s3) * s_lut[2 * q + 1];
    float l2[4];
#pragma unroll
    for (int q = 0; q < 4; ++q)
        l2[q] = s2 * l3[2 * q] + (1.0f - s2) * l3[2 * q + 1];
    float l1[2];
#pragma unroll
    for (int q = 0; q < 2; ++q)
        l1[q] = s1 * l2[2 * q] + (1.0f - s1) * l2[2 * q + 1];
    const float lut_out = s0 * l1[0] + (1.0f - s0) * l1[1];

    out[blockRow0 + t] = s_mix[0] * lut_out + s_mix[1] * a1 + s_mix[2] * a2;
}

extern "C" void kernel_launch(void* const* d_in, const int* in_sizes, int n_in,
                              void* d_out, int out_size, void* d_ws, size_t ws_size,
                              hipStream_t stream) {
    const float* x    = (const float*)d_in[0];  // [B, 64]
    const float* cp   = (const float*)d_in[1];  // [64, 4]
    const float* lut  = (const float*)d_in[2];  // [2,2,2,2]
    const float* mixp = (const float*)d_in[3];  // [3]
    float* out = (float*)d_out;                 // [B]
    float* ws  = (float*)d_ws;

    const int Btot   = in_sizes[0] / C_IN;      // 1048576
    const int blocks = (Btot + 255) / 256;

    LAB_42906723287350_prep<<<1, 256, 0, stream>>>(cp, lut, mixp, ws);
    LAB_42906723287350_main<<<blocks, 256, 0, stream>>>(x, ws, out);
}